// GTLayer_88450556494512
// MI455X (gfx1250) — compile-verified
//
#include <hip/hip_runtime.h>
#include <hip/hip_bf16.h>

typedef float v2f __attribute__((ext_vector_type(2)));
typedef float v8f __attribute__((ext_vector_type(8)));

#define DD 64   // embedding size
#define HH 4    // heads
#define DH 16   // dim per head

// ---------------------------------------------------------------------------
// Phase 0a: zero the output accumulator and the softmax-denominator buffer.
// ---------------------------------------------------------------------------
__global__ void zero2_kernel(float* __restrict__ a, int na,
                             float* __restrict__ b, int nb) {
    int i = blockIdx.x * blockDim.x + threadIdx.x;
    int stride = gridDim.x * blockDim.x;
    for (int j = i; j < na; j += stride) a[j] = 0.0f;
    for (int j = i; j < nb; j += stride) b[j] = 0.0f;
}

// ---------------------------------------------------------------------------
// Phase 0b: transpose the three 64x64 weights into WT[w][n][k] = W_w[k][n]
// so the WMMA B fragment (two consecutive k values for a fixed column n)
// becomes a single contiguous global_load_b64.
// ---------------------------------------------------------------------------
__global__ void transpose_w_kernel(const float* __restrict__ Wq,
                                   const float* __restrict__ Wk,
                                   const float* __restrict__ Wv,
                                   float* __restrict__ WT) {
    int idx = blockIdx.x * blockDim.x + threadIdx.x;   // 3*64*64 = 12288
    if (idx >= 3 * DD * DD) return;
    int w = idx / (DD * DD);
    int r = idx % (DD * DD);
    int n = r / DD;
    int k = r % DD;
    const float* W = (w == 0) ? Wq : ((w == 1) ? Wk : Wv);
    WT[idx] = W[k * DD + n];
}

// ---------------------------------------------------------------------------
// Phase 1: node-level projections  Q = X@Wq, K = X@Wk, V = X@Wv  via
// V_WMMA_F32_16X16X4_F32. One wave handles a 16-row tile of X; the A
// fragments (16 k-steps x v2f = 32 VGPRs) are loaded once and reused for all
// 3 weight matrices x 4 N-tiles (192 WMMAs per tile).
//
// A layout (32-bit A 16x4, ISA 7.12.2): lane l<16 holds X[m][k..k+1],
// lane l>=16 holds X[m][k+2..k+3], m = l&15.
// B fragment from transposed weights: b = WT[w][n0+m][kb..kb+1] (b64 load).
// D layout: VGPR i -> row (mbase+i), col (lane&15), mbase = (lane>=16)*8.
// Store guard is hoisted to a uniform per-tile branch so the hot path has
// no exec-mask manipulation.
// ---------------------------------------------------------------------------
__global__ void __launch_bounds__(256) gemm_qkv_kernel(
    const float* __restrict__ X,
    const float* __restrict__ WT,   // [3][64][64] transposed weights
    float* __restrict__ Qo, float* __restrict__ Ko, float* __restrict__ Vo,
    int N, int tiles) {
    const int wave = threadIdx.x >> 5;
    const int lane = threadIdx.x & 31;
    const int tile = blockIdx.x * 8 + wave;
    if (tile >= tiles) return;

    const int row0  = tile * 16;
    const int m     = lane & 15;
    const int khalf = (lane >> 4) << 1;   // 0 or 2
    const int mbase = (lane >> 4) << 3;   // 0 or 8

    int arow = row0 + m;
    if (arow >= N) arow = N - 1;          // clamp (EXEC must stay all-ones)
    const float* xrow = X + (size_t)arow * DD;

    // A fragments for all 16 k-steps (K = 64, 4 per WMMA), reused 12x.
    v2f a[16];
#pragma unroll
    for (int s = 0; s < 16; ++s)
        a[s] = *(const v2f*)(xrow + 4 * s + khalf);

    float* Os[3] = {Qo, Ko, Vo};
    const bool full = (row0 + 16 <= N);

#pragma unroll
    for (int w = 0; w < 3; ++w) {
        const float* Wt = WT + w * DD * DD;
        float*       O  = Os[w];
#pragma unroll
        for (int nt = 0; nt < 4; ++nt) {
            const int n0 = nt * 16;
            // Lane m's B column: WT row (n0+m), k offset khalf.
            const float* wb = Wt + (n0 + m) * DD + khalf;
            v8f acc = {0.f, 0.f, 0.f, 0.f, 0.f, 0.f, 0.f, 0.f};
#pragma unroll
            for (int s = 0; s < 16; ++s) {
                v2f b = *(const v2f*)(wb + 4 * s);
                acc = __builtin_amdgcn_wmma_f32_16x16x4_f32(
                    /*neg_a=*/false, a[s], /*neg_b=*/false, b,
                    /*c_mod=*/(short)0, acc,
                    /*reuse_a=*/false, /*reuse_b=*/false);
            }
            float* obase = O + (size_t)(row0 + mbase) * DD + n0 + m;
            if (full) {
#pragma unroll
                for (int i = 0; i < 8; ++i)
                    obase[i * DD] = acc[i];
            } else {
#pragma unroll
                for (int i = 0; i < 8; ++i)
                    if (row0 + mbase + i < N) obase[i * DD] = acc[i];
            }
        }
    }
}

// ---------------------------------------------------------------------------
// Phase 2: per-(edge, head) attention score.
//   s = dot16(Q[row][h], K[col][h]); clip(-10,10); e = exp(s)
//   expatt[e,h] = e;  norm[row,h] += e  (global_atomic_add_f32)
// ---------------------------------------------------------------------------
__global__ void __launch_bounds__(256) edge_score_kernel(
    const float* __restrict__ Q, const float* __restrict__ K,
    const int* __restrict__ rows, const int* __restrict__ cols,
    float* __restrict__ expatt, float* __restrict__ norm, int EH) {
    int idx = blockIdx.x * blockDim.x + threadIdx.x;
    if (idx >= EH) return;
    const int e = idx >> 2;
    const int h = idx & 3;
    const int r = rows[e];
    const int c = cols[e];

    const float4* q4 = (const float4*)(Q + (size_t)r * DD + h * DH);
    const float4* k4 = (const float4*)(K + (size_t)c * DD + h * DH);
    float s = 0.0f;
#pragma unroll
    for (int i = 0; i < 4; ++i) {
        float4 q = q4[i];
        float4 k = k4[i];
        s += q.x * k.x + q.y * k.y + q.z * k.z + q.w * k.w;
    }
    s = fminf(10.0f, fmaxf(-10.0f, s));
    float ex = __expf(s);
    expatt[idx] = ex;
    unsafeAtomicAdd(&norm[(size_t)r * HH + h], ex);
}

// ---------------------------------------------------------------------------
// Phase 3: per-(edge, head) weighted scatter-add.
//   w = expatt[e,h] / (norm[row,h] + 1e-8)
//   out[row][h*16 .. h*16+15] += w * V[col][h*16 .. h*16+15]
// ---------------------------------------------------------------------------
__global__ void __launch_bounds__(256) edge_aggregate_kernel(
    const float* __restrict__ V,
    const int* __restrict__ rows, const int* __restrict__ cols,
    const float* __restrict__ expatt, const float* __restrict__ norm,
    float* __restrict__ out, int EH) {
    int idx = blockIdx.x * blockDim.x + threadIdx.x;
    if (idx >= EH) return;
    const int e = idx >> 2;
    const int h = idx & 3;
    const int r = rows[e];
    const int c = cols[e];

    const float w = expatt[idx] / (norm[(size_t)r * HH + h] + 1e-8f);
    const float4* v4 = (const float4*)(V + (size_t)c * DD + h * DH);
    float* o = out + (size_t)r * DD + h * DH;
#pragma unroll
    for (int i = 0; i < 4; ++i) {
        float4 v = v4[i];
        unsafeAtomicAdd(o + 4 * i + 0, w * v.x);
        unsafeAtomicAdd(o + 4 * i + 1, w * v.y);
        unsafeAtomicAdd(o + 4 * i + 2, w * v.z);
        unsafeAtomicAdd(o + 4 * i + 3, w * v.w);
    }
}

// ---------------------------------------------------------------------------
extern "C" void kernel_launch(void* const* d_in, const int* in_sizes, int n_in,
                              void* d_out, int out_size, void* d_ws, size_t ws_size,
                              hipStream_t stream) {
    const float* emb = (const float*)d_in[0];
    const float* Wq  = (const float*)d_in[1];
    const float* Wk  = (const float*)d_in[2];
    const float* Wv  = (const float*)d_in[3];
    const int*   ei  = (const int*)d_in[4];

    const int N = in_sizes[0] / DD;      // 100000
    const int E = in_sizes[4] / 2;       // 800000
    const int* rows = ei;
    const int* cols = ei + E;

    float* out = (float*)d_out;

    // Workspace layout (floats): Q | K | V | norm | expatt | WT
    float* ws = (float*)d_ws;
    const size_t ND = (size_t)N * DD;
    float* Qd      = ws;
    float* Kd      = ws + ND;
    float* Vd      = ws + 2 * ND;
    float* normd   = ws + 3 * ND;                         // N*HH
    float* expattd = ws + 3 * ND + (size_t)N * HH;        // E*HH
    float* WTd     = expattd + (size_t)E * HH;            // 3*64*64

    // Phase 0: zero out + norm; transpose weights for b64 B-fragment loads.
    zero2_kernel<<<1024, 256, 0, stream>>>(out, N * DD, normd, N * HH);
    transpose_w_kernel<<<(3 * DD * DD + 255) / 256, 256, 0, stream>>>(
        Wq, Wk, Wv, WTd);

    // Phase 1: Q/K/V node projections via f32 WMMA.
    const int tiles = (N + 15) / 16;
    gemm_qkv_kernel<<<(tiles + 7) / 8, 256, 0, stream>>>(
        emb, WTd, Qd, Kd, Vd, N, tiles);

    // Phase 2: edge scores + segment softmax denominator.
    const int EH = E * HH;
    edge_score_kernel<<<(EH + 255) / 256, 256, 0, stream>>>(
        Qd, Kd, rows, cols, expattd, normd, EH);

    // Phase 3: normalized weighted scatter-add into out.
    edge_aggregate_kernel<<<(EH + 255) / 256, 256, 0, stream>>>(
        Vd, rows, cols, expattd, normd, out, EH);
}